// AssociativeMemory_54339926229372
// MI455X (gfx1250) — compile-verified
//
#include <hip/hip_runtime.h>
#include <math.h>

// ---------------------------------------------------------------------------
// Problem constants (from reference): B=16384, S=32, D=64, K=3
// ---------------------------------------------------------------------------
#define NB     16384
#define NS     32
#define ND     64
#define EPS_LN 1e-5f

typedef __attribute__((ext_vector_type(2))) float v2f;
typedef __attribute__((ext_vector_type(8))) float v8f;

// ---------------------------------------------------------------------------
// Wave-level helpers (wave32)
// ---------------------------------------------------------------------------
__device__ __forceinline__ float waveAllSum(float v) {
#pragma unroll
  for (int o = 16; o > 0; o >>= 1) v += __shfl_xor(v, o, 32);
  return v;
}

__device__ __forceinline__ float waveAllMax(float v) {
#pragma unroll
  for (int o = 16; o > 0; o >>= 1) v = fmaxf(v, __shfl_xor(v, o, 32));
  return v;
}

// Reduce 32 per-lane partials part[s] (s = slot) across all 32 lanes.
// On return lane l holds the full sum of part[l] over all lanes.
__device__ __forceinline__ float transposeReduce32(float (&p)[32], int lane) {
#pragma unroll
  for (int o = 16; o >= 1; o >>= 1) {
#pragma unroll
    for (int k = 0; k < o; ++k) {
      float a  = p[k];
      float c  = p[k + o];
      float oa = __shfl_xor(a, o, 32);
      float oc = __shfl_xor(c, o, 32);
      p[k] = (lane & o) ? (c + oc) : (a + oa);
    }
  }
  return p[0];
}

// ---------------------------------------------------------------------------
// Kernel A: logits = flat(B,128) @ Wa(128,32) + ba  via V_WMMA_F32_16X16X4_F32
//           gate   = sigmoid(flat @ Wg + bg)
// Each wave handles 16 consecutive batches (the WMMA M dimension).
// Block = 256 threads = 8 waves = 128 batches/block; grid = 128 blocks.
// ---------------------------------------------------------------------------
__global__ __launch_bounds__(256) void AM_logits_wmma_kernel(
    const float* __restrict__ gw_r, const float* __restrict__ gw_i,
    const float* __restrict__ Wg,   const float* __restrict__ bg,
    const float* __restrict__ Wa,   const float* __restrict__ ba,
    float* __restrict__ ws_logits,  float* __restrict__ ws_gate) {
  const int lane = threadIdx.x & 31;
  const int wv   = threadIdx.x >> 5;
  const int base = (blockIdx.x * 8 + wv) * 16;  // first batch of this wave tile
  const int m    = lane & 15;                   // A-row / B-col within tile
  const int hi   = lane >> 4;                   // which half-wave
  const int koff = hi << 1;                     // K offset: 0 or 2
  const int bA   = base + m;                    // batch carried by this lane (A row)

  v8f acc0 = {0.f, 0.f, 0.f, 0.f, 0.f, 0.f, 0.f, 0.f};  // N = 0..15
  v8f acc1 = {0.f, 0.f, 0.f, 0.f, 0.f, 0.f, 0.f, 0.f};  // N = 16..31

#pragma unroll
  for (int k0 = 0; k0 < 128; k0 += 4) {
    const int kk = k0 + koff;
    // A operand: flat[b][k] = k<64 ? gw_r[b][k] : gw_i[b][k-64]
    // A 16x4 layout: lane m holds row M=m; VGPR0/1 = K = koff, koff+1
    const float* asrc =
        (kk < 64) ? (gw_r + (size_t)bA * ND + kk)
                  : (gw_i + (size_t)bA * ND + (kk - 64));
    v2f a = *(const v2f*)asrc;

    // B operand 4x16: VGPR0 lanes0-15 = row K=k0 (N=0..15), lanes16-31 row K=k0+2
    v2f b0, b1;
    b0.x = Wa[(size_t)kk * NS + m];
    b0.y = Wa[(size_t)(kk + 1) * NS + m];
    b1.x = Wa[(size_t)kk * NS + 16 + m];
    b1.y = Wa[(size_t)(kk + 1) * NS + 16 + m];

    acc0 = __builtin_amdgcn_wmma_f32_16x16x4_f32(false, a, false, b0,
                                                 (short)0, acc0, false, false);
    acc1 = __builtin_amdgcn_wmma_f32_16x16x4_f32(false, a, false, b1,
                                                 (short)0, acc1, false, false);
  }

  // C layout: VGPR j -> row M = j + 8*hi, col N = m (tile0) / 16+m (tile1)
#pragma unroll
  for (int j = 0; j < 8; ++j) {
    const int M = j + 8 * hi;
    ws_logits[(size_t)(base + M) * NS + m]      = acc0[j] + ba[m];
    ws_logits[(size_t)(base + M) * NS + 16 + m] = acc1[j] + ba[16 + m];
  }

  // Write gate: lane pair (m, m+16) split the 128-long dot product.
  {
    const float* src  = hi ? (gw_i + (size_t)bA * ND) : (gw_r + (size_t)bA * ND);
    const float* wsrc = Wg + hi * 64;
    float gp = 0.f;
#pragma unroll
    for (int k = 0; k < 64; ++k) gp += src[k] * wsrc[k];
    gp += __shfl_xor(gp, 16, 32);
    if (hi == 0) ws_gate[bA] = 1.f / (1.f + expf(-(gp + bg[0])));
  }
}

// ---------------------------------------------------------------------------
// Kernel B: fully fused streaming pass. One wave per batch, lane owns
// columns {2l, 2l+1}. prev_mem rows live in registers; all global traffic
// is coalesced 256B b64 transactions. Block = 128 threads (4 batches).
// ---------------------------------------------------------------------------
__global__ __launch_bounds__(128) void AM_fused_kernel(
    const float* __restrict__ gw_r, const float* __restrict__ gw_i,
    const float* __restrict__ pm_r, const float* __restrict__ pm_i,
    const float* __restrict__ gamma_r, const float* __restrict__ beta_r,
    const float* __restrict__ gamma_i, const float* __restrict__ beta_i,
    const float* __restrict__ ws_logits, const float* __restrict__ ws_gate,
    float* __restrict__ o_read_r, float* __restrict__ o_read_i,
    float* __restrict__ o_next_r, float* __restrict__ o_next_i,
    float* __restrict__ ws_entp) {
  const int lane = threadIdx.x & 31;
  const int w    = threadIdx.x >> 5;
  const int b    = blockIdx.x * 4 + w;
  const int c0   = lane << 1;                   // first owned column
  const size_t gb = (size_t)b * ND;
  const size_t mb = (size_t)b * (NS * ND);

  const v2f gr = *(const v2f*)(gw_r + gb + c0);
  const v2f gi = *(const v2f*)(gw_i + gb + c0);

  v2f mr[NS], mi[NS];
#pragma unroll
  for (int s = 0; s < NS; ++s) {
    mr[s] = *(const v2f*)(pm_r + mb + s * ND + c0);
    mi[s] = *(const v2f*)(pm_i + mb + s * ND + c0);
  }

  // ---- sim[s] = <mem_r[s], gw_r> + <mem_i[s], gw_i> -----------------------
  float part[NS];
#pragma unroll
  for (int s = 0; s < NS; ++s)
    part[s] = mr[s].x * gr.x + mr[s].y * gr.y + mi[s].x * gi.x + mi[s].y * gi.y;
  const float sim = transposeReduce32(part, lane);  // lane l holds sim[l]

  // ---- attn = softmax(sim) ------------------------------------------------
  const float smax = waveAllMax(sim);
  const float sexp = expf(sim - smax);
  const float ssum = waveAllSum(sexp);
  const float attn = sexp / ssum;

  // ---- read = attn @ mem --------------------------------------------------
  v2f rr = {0.f, 0.f}, ri = {0.f, 0.f};
#pragma unroll
  for (int s = 0; s < NS; ++s) {
    const float a = __shfl(attn, s, 32);
    rr.x += a * mr[s].x; rr.y += a * mr[s].y;
    ri.x += a * mi[s].x; ri.y += a * mi[s].y;
  }
  *(v2f*)(o_read_r + gb + c0) = rr;
  *(v2f*)(o_read_i + gb + c0) = ri;

  // ---- slot weights ww = softmax(logits) (logits from WMMA kernel) --------
  const float lg   = ws_logits[(size_t)b * NS + lane];
  const float lmax = waveAllMax(lg);
  const float lexp = expf(lg - lmax);
  const float lsum = waveAllSum(lexp);
  const float ww   = lexp / lsum;

  // ---- entropy (deterministic fixed-order block partial) ------------------
  const float ent = waveAllSum(-ww * logf(ww + 1e-10f));
  __shared__ float entsh[4];
  if (lane == 0) entsh[w] = ent;
  __syncthreads();
  if (threadIdx.x == 0)
    ws_entp[blockIdx.x] = ((entsh[0] + entsh[1]) + entsh[2]) + entsh[3];

  // ---- top-3 with lowest-index tie-break (matches lax.top_k) --------------
  float wv0 = ww;
  float tv[3];
  int   ti[3];
#pragma unroll
  for (int t = 0; t < 3; ++t) {
    float v = wv0;
    int   i = lane;
#pragma unroll
    for (int o = 16; o > 0; o >>= 1) {
      const float ov = __shfl_xor(v, o, 32);
      const int   oi = __shfl_xor(i, o, 32);
      if (ov > v || (ov == v && oi < i)) { v = ov; i = oi; }
    }
    tv[t] = v;
    ti[t] = i;
    if (lane == i) wv0 = -3.402823466e38f;
  }
  const float tsum   = tv[0] + tv[1] + tv[2] + 1e-6f;
  const float sparse = (lane == ti[0]) ? tv[0]
                     : (lane == ti[1]) ? tv[1]
                     : (lane == ti[2]) ? tv[2] : 0.f;
  const float gate = ws_gate[b];
  const float eff  = gate * (sparse / tsum);  // eff for slot s = lane

  // ---- next_* candidate + LayerNorm stats (row stats via transpose-reduce)
  float Sp[NS], Qp[NS];
#pragma unroll
  for (int s = 0; s < NS; ++s) {
    const float e  = __shfl(eff, s, 32);
    const float vx = (1.f - e) * mr[s].x + e * gr.x;
    const float vy = (1.f - e) * mr[s].y + e * gr.y;
    Sp[s] = vx + vy;
    Qp[s] = vx * vx + vy * vy;
  }
  const float Sr     = transposeReduce32(Sp, lane);
  const float Qr     = transposeReduce32(Qp, lane);
  const float mean_r = Sr * (1.f / 64.f);
  const float rstd_r = rsqrtf(Qr * (1.f / 64.f) - mean_r * mean_r + EPS_LN);

#pragma unroll
  for (int s = 0; s < NS; ++s) {
    const float e  = __shfl(eff, s, 32);
    const float vx = (1.f - e) * mi[s].x + e * gi.x;
    const float vy = (1.f - e) * mi[s].y + e * gi.y;
    Sp[s] = vx + vy;
    Qp[s] = vx * vx + vy * vy;
  }
  const float Si     = transposeReduce32(Sp, lane);
  const float Qi     = transposeReduce32(Qp, lane);
  const float mean_i = Si * (1.f / 64.f);
  const float rstd_i = rsqrtf(Qi * (1.f / 64.f) - mean_i * mean_i + EPS_LN);

  const v2f gmr = *(const v2f*)(gamma_r + c0);
  const v2f btr = *(const v2f*)(beta_r + c0);
  const v2f gmi = *(const v2f*)(gamma_i + c0);
  const v2f bti = *(const v2f*)(beta_i + c0);

  // ---- normalize + coalesced store ---------------------------------------
#pragma unroll
  for (int s = 0; s < NS; ++s) {
    const float e   = __shfl(eff, s, 32);
    const float m_r = __shfl(mean_r, s, 32);
    const float r_r = __shfl(rstd_r, s, 32);
    const float m_i = __shfl(mean_i, s, 32);
    const float r_i = __shfl(rstd_i, s, 32);

    const float vx = (1.f - e) * mr[s].x + e * gr.x;
    const float vy = (1.f - e) * mr[s].y + e * gr.y;
    v2f outr;
    outr.x = (vx - m_r) * r_r * gmr.x + btr.x;
    outr.y = (vy - m_r) * r_r * gmr.y + btr.y;
    *(v2f*)(o_next_r + mb + s * ND + c0) = outr;

    const float ux = (1.f - e) * mi[s].x + e * gi.x;
    const float uy = (1.f - e) * mi[s].y + e * gi.y;
    v2f outi;
    outi.x = (ux - m_i) * r_i * gmi.x + bti.x;
    outi.y = (uy - m_i) * r_i * gmi.y + bti.y;
    *(v2f*)(o_next_i + mb + s * ND + c0) = outi;
  }
}

// ---------------------------------------------------------------------------
// Kernel C: deterministic reduction of per-block entropy partials.
// ---------------------------------------------------------------------------
__global__ __launch_bounds__(256) void AM_entropy_reduce_kernel(
    const float* __restrict__ partials, float* __restrict__ o_ent, int n) {
  __shared__ float sh[256];
  float a = 0.f;
  for (int i = threadIdx.x; i < n; i += 256) a += partials[i];
  sh[threadIdx.x] = a;
  __syncthreads();
  for (int o = 128; o > 0; o >>= 1) {
    if ((int)threadIdx.x < o) sh[threadIdx.x] += sh[threadIdx.x + o];
    __syncthreads();
  }
  if (threadIdx.x == 0) o_ent[0] = sh[0] * (1.f / (float)NB);
}

// ---------------------------------------------------------------------------
// Host launcher
// ---------------------------------------------------------------------------
extern "C" void kernel_launch(void* const* d_in, const int* in_sizes, int n_in,
                              void* d_out, int out_size, void* d_ws,
                              size_t ws_size, hipStream_t stream) {
  (void)in_sizes; (void)n_in; (void)out_size; (void)ws_size;

  const float* gw_r    = (const float*)d_in[0];
  const float* gw_i    = (const float*)d_in[1];
  const float* pm_r    = (const float*)d_in[2];
  const float* pm_i    = (const float*)d_in[3];
  const float* Wg      = (const float*)d_in[4];
  const float* bg      = (const float*)d_in[5];
  const float* Wa      = (const float*)d_in[6];
  const float* ba      = (const float*)d_in[7];
  const float* gamma_r = (const float*)d_in[8];
  const float* beta_r  = (const float*)d_in[9];
  const float* gamma_i = (const float*)d_in[10];
  const float* beta_i  = (const float*)d_in[11];

  float* out      = (float*)d_out;
  float* o_read_r = out;
  float* o_read_i = out + (size_t)NB * ND;
  float* o_next_r = out + (size_t)2 * NB * ND;
  float* o_next_i = o_next_r + (size_t)NB * NS * ND;
  float* o_ent    = o_next_i + (size_t)NB * NS * ND;

  float* ws        = (float*)d_ws;
  float* ws_logits = ws;                             // NB*NS floats
  float* ws_gate   = ws_logits + (size_t)NB * NS;    // NB floats
  float* ws_entp   = ws_gate + NB;                   // 4096 floats

  // A: WMMA logits + gate. 128 blocks x 256 threads (8 waves x 16 batches).
  AM_logits_wmma_kernel<<<NB / 128, 256, 0, stream>>>(
      gw_r, gw_i, Wg, bg, Wa, ba, ws_logits, ws_gate);

  // B: fused streaming pass. 4096 blocks x 128 threads (4 batches/block).
  AM_fused_kernel<<<NB / 4, 128, 0, stream>>>(
      gw_r, gw_i, pm_r, pm_i, gamma_r, beta_r, gamma_i, beta_i,
      ws_logits, ws_gate, o_read_r, o_read_i, o_next_r, o_next_i, ws_entp);

  // C: deterministic entropy mean.
  AM_entropy_reduce_kernel<<<1, 256, 0, stream>>>(ws_entp, o_ent, NB / 4);
}